// TEColumnParallelLinear_25958782337557
// MI455X (gfx1250) — compile-verified
//
#include <hip/hip_runtime.h>
#include <stdint.h>

typedef __attribute__((ext_vector_type(2))) float v2f;
typedef __attribute__((ext_vector_type(8))) float v8f;

// Problem sizes (fixed by the reference)
constexpr int SEQ = 4096, BATCH = 2, D_IN = 4096, D_OUT = 16384;
constexpr int M = SEQ * BATCH;   // 8192
constexpr int N = D_OUT;         // 16384
constexpr int K = D_IN;          // 4096

// Tiling
constexpr int BM = 128;
constexpr int BN = 128;
constexpr int BK = 32;
constexpr int LDS_STRIDE = BK + 4;              // 36 floats; 144B row = 16B-aligned
constexpr int TILE_FLOATS = BM * LDS_STRIDE;    // 4608 floats / 18432 B per tile
constexpr int STAGE_FLOATS = 2 * TILE_FLOATS;   // A+B per stage
constexpr int STAGE_BYTES = STAGE_FLOATS * 4;   // 36864 B
constexpr int MB = M / BM;                      // 64
constexpr int NB = N / BN;                      // 128
constexpr int PANEL = 16;                       // N-blocks per L2 panel (32 MiB weight)

__device__ __forceinline__ void async_copy16(uint32_t lds_off, const float* g) {
  // GLOBAL_LOAD_ASYNC_TO_LDS_B128: vdst = LDS byte address, vaddr = 64-bit global
  asm volatile("global_load_async_to_lds_b128 %0, %1, off"
               :: "v"(lds_off), "v"((uint64_t)(uintptr_t)g)
               : "memory");
}

__device__ __forceinline__ void wait_async0() {
  asm volatile("s_wait_asynccnt 0x0" ::: "memory");
}

__global__ __launch_bounds__(256) void gemm_bias_wmma_f32(
    const float* __restrict__ A,    // x       [M, K]
    const float* __restrict__ W,    // weight  [N, K]
    const float* __restrict__ bias, // [N]
    float* __restrict__ C)          // out     [M, N]
{
  __shared__ float lds[2 * STAGE_FLOATS];  // [stage][A|B][row][LDS_STRIDE]

  // ---- L2-panel block swizzle: panels of PANEL n-blocks, M-fastest inside ----
  const int id = blockIdx.x;
  constexpr int PER_PANEL = MB * PANEL;     // 1024 blocks per panel
  const int panel = id / PER_PANEL;
  const int rem   = id % PER_PANEL;
  const int nBlock = (panel * PANEL + rem / MB) * BN;
  const int mBlock = (rem % MB) * BM;

  const int tid  = threadIdx.x;
  const int lane = tid & 31;
  const int wave = tid >> 5;
  const int wm   = wave >> 2;    // 0..1
  const int wn   = wave & 3;     // 0..3

  // ---- per-thread async-copy addressing: 4 rows of A + 4 rows of B, b128 each ----
  const int loadRow = tid >> 3;        // 0..31
  const int loadCol = (tid & 7) * 4;   // 0..28 (floats)

  const uint32_t ldsBase = (uint32_t)(uintptr_t)lds;  // generic LDS ptr low 32b = LDS offset
  uint32_t ldsA[4], ldsB[4];
  const float* gA[4];
  const float* gB[4];
  #pragma unroll
  for (int i = 0; i < 4; ++i) {
    const int row = loadRow + i * 32;
    ldsA[i] = ldsBase + (uint32_t)((row * LDS_STRIDE + loadCol) * 4);
    ldsB[i] = ldsA[i] + (uint32_t)(TILE_FLOATS * 4);
    gA[i] = A + (size_t)(mBlock + row) * K + loadCol;
    gB[i] = W + (size_t)(nBlock + row) * K + loadCol;
  }

  // ---- prologue: stage 0 ----
  #pragma unroll
  for (int i = 0; i < 4; ++i) {
    async_copy16(ldsA[i], gA[i]);
    async_copy16(ldsB[i], gB[i]);
    gA[i] += BK;
    gB[i] += BK;
  }
  wait_async0();
  __syncthreads();

  v8f acc[4][2] = {};
  const int r15 = lane & 15;
  const int kHi = (lane >> 4) << 1;   // lanes 16..31 read k+2 (f32 WMMA A/B layout)

  int stage = 0;
  for (int k0 = 0; k0 < K; k0 += BK) {
    const int next = stage ^ 1;

    // issue next tile's async copies before computing on the current one
    if (k0 + BK < K) {
      const uint32_t soff = (uint32_t)(next * STAGE_BYTES);
      #pragma unroll
      for (int i = 0; i < 4; ++i) {
        async_copy16(ldsA[i] + soff, gA[i]);
        async_copy16(ldsB[i] + soff, gB[i]);
        gA[i] += BK;
        gB[i] += BK;
      }
    }

    const float* As = lds + stage * STAGE_FLOATS;
    const float* Bs = As + TILE_FLOATS;

    // ---- 8 k-steps of V_WMMA_F32_16X16X4_F32 on the current stage ----
    #pragma unroll
    for (int kk = 0; kk < BK; kk += 4) {
      const int kA = kk + kHi;
      v2f a[4], b[2];
      #pragma unroll
      for (int i = 0; i < 4; ++i)
        a[i] = *(const v2f*)&As[(wm * 64 + i * 16 + r15) * LDS_STRIDE + kA];
      #pragma unroll
      for (int j = 0; j < 2; ++j)
        b[j] = *(const v2f*)&Bs[(wn * 32 + j * 16 + r15) * LDS_STRIDE + kA];

      #pragma unroll
      for (int i = 0; i < 4; ++i)
        #pragma unroll
        for (int j = 0; j < 2; ++j)
          acc[i][j] = __builtin_amdgcn_wmma_f32_16x16x4_f32(
              false, a[i], false, b[j], (short)0, acc[i][j], false, false);
    }

    wait_async0();      // next stage's data landed in LDS
    __syncthreads();    // all waves done reading current stage
    stage = next;
  }

  // ---- epilogue: bias add + non-temporal stores (output is write-once) ----
  const int rHi = (lane >> 4) << 3;  // 0 or 8
  #pragma unroll
  for (int j = 0; j < 2; ++j) {
    const int col = nBlock + wn * 32 + j * 16 + r15;
    const float bv = bias[col];
    #pragma unroll
    for (int i = 0; i < 4; ++i) {
      const int rowBase = mBlock + wm * 64 + i * 16 + rHi;
      #pragma unroll
      for (int r = 0; r < 8; ++r) {
        __builtin_nontemporal_store(acc[i][j][r] + bv,
                                    &C[(size_t)(rowBase + r) * N + col]);
      }
    }
  }
}

extern "C" void kernel_launch(void* const* d_in, const int* in_sizes, int n_in,
                              void* d_out, int out_size, void* d_ws, size_t ws_size,
                              hipStream_t stream) {
  const float* x  = (const float*)d_in[0];   // [SEQ, BATCH, D_IN] -> [M, K]
  const float* w  = (const float*)d_in[1];   // [D_OUT, D_IN]      -> [N, K]
  const float* b  = (const float*)d_in[2];   // [D_OUT]
  float* out      = (float*)d_out;           // [M, N]

  gemm_bias_wmma_f32<<<dim3(MB * NB, 1, 1), dim3(256, 1, 1), 0, stream>>>(x, w, b, out);
}